// PaDiM_16312285790243
// MI455X (gfx1250) — compile-verified
//
#include <hip/hip_runtime.h>

typedef __attribute__((ext_vector_type(2))) float v2f;
typedef __attribute__((ext_vector_type(8))) float v8f;

#define BB 32    // batch
#define CC 192   // channels
#define PP 3136  // patches
// LDS row stride: 192 + 8. Makes the two half-wave (lanes 0-15 vs 16-31)
// operand fetches hit disjoint 16-bank groups of the 64 LDS banks
// (2*200 mod 64 == 16), avoiding 2-way conflicts on every WMMA operand load.
#define LDS_STRIDE 200

__global__ __launch_bounds__(256) void padim_gram_kernel(
    const float* __restrict__ emb,      // [B, C, P]
    const float* __restrict__ means_in, // [P, C]
    const float* __restrict__ cov_in,   // [P, C, C]
    float* __restrict__ out_means,      // [P, C]
    float* __restrict__ out_cov) {      // [P, C, C]
  __shared__ float lds[BB * LDS_STRIDE];

  const int tid   = threadIdx.x;
  const int lane  = tid & 31;
  const int wave  = tid >> 5;    // 8 waves per workgroup
  const int half  = lane >> 4;   // 0: lanes 0-15, 1: lanes 16-31
  const int l     = lane & 15;
  const int patch = blockIdx.x;

  // ---- Stage the 32x192 patch slice into LDS -------------------------------
  // emb element (b, c) of this patch sits at b*C*P + c*P + patch (stride-P in
  // c). Uncoalesced per-lane, but the full embeddings tensor (77 MB) fits in
  // the 192 MB L2, and adjacent patch-workgroups consume adjacent bytes of
  // each line, so HBM sees a clean stream.
  for (int idx = tid; idx < BB * CC; idx += 256) {
    const int b = idx / CC;
    const int c = idx - b * CC;
    lds[b * LDS_STRIDE + c] =
        emb[(size_t)b * (CC * PP) + (size_t)c * PP + (size_t)patch];
  }
  __syncthreads();

  // ---- Means: column sums over the batch ----------------------------------
  for (int c = tid; c < CC; c += 256) {
    float s = 0.0f;
#pragma unroll
    for (int b = 0; b < BB; ++b) s += lds[b * LDS_STRIDE + c];
    const size_t o = (size_t)patch * CC + (size_t)c;
    out_means[o] = means_in[o] + s;
  }

  // ---- Gram matrix: 144 16x16 tiles, 18 per wave (uniform -> EXEC all-1s) --
  const size_t covbase = (size_t)patch * (CC * CC);
  for (int tile = wave; tile < 144; tile += 8) {
    const int tm  = tile / 12;
    const int tn  = tile - tm * 12;
    const int col = tn * 16 + l;

    // C/D 16x16 f32 layout: VGPR r -> row r (lanes 0-15) / row r+8 (16-31).
    v8f acc;
#pragma unroll
    for (int r = 0; r < 8; ++r) {
      const int row = tm * 16 + r + half * 8;
      acc[r] = __builtin_nontemporal_load(
          &cov_in[covbase + (size_t)row * CC + (size_t)col]);
    }

    // K = batch = 32, in chunks of 4 per v_wmma_f32_16x16x4_f32.
    // A (16x4 f32): lanes 0-15 hold K=k0,k0+1; lanes 16-31 hold K=k0+2,k0+3.
    // B (4x16 f32): rows striped across lanes with the same half-wave split.
#pragma unroll
    for (int k0 = 0; k0 < BB; k0 += 4) {
      const int ka = k0 + half * 2;
      v2f a, b;
      a.x = lds[ka * LDS_STRIDE + tm * 16 + l];
      a.y = lds[(ka + 1) * LDS_STRIDE + tm * 16 + l];
      b.x = lds[ka * LDS_STRIDE + col];
      b.y = lds[(ka + 1) * LDS_STRIDE + col];
      acc = __builtin_amdgcn_wmma_f32_16x16x4_f32(
          /*neg_a=*/false, a, /*neg_b=*/false, b,
          /*c_mod=*/(short)0, acc, /*reuse_a=*/false, /*reuse_b=*/false);
    }

#pragma unroll
    for (int r = 0; r < 8; ++r) {
      const int row = tm * 16 + r + half * 8;
      __builtin_nontemporal_store(
          acc[r], &out_cov[covbase + (size_t)row * CC + (size_t)col]);
    }
  }
}

extern "C" void kernel_launch(void* const* d_in, const int* in_sizes, int n_in,
                              void* d_out, int out_size, void* d_ws, size_t ws_size,
                              hipStream_t stream) {
  const float* emb      = (const float*)d_in[0];  // [32, 192, 3136]
  const float* means_in = (const float*)d_in[1];  // [3136, 192]
  const float* cov_in   = (const float*)d_in[2];  // [3136, 192, 192]

  float* out_means = (float*)d_out;                       // [3136, 192]
  float* out_cov   = out_means + (size_t)PP * CC;         // [3136, 192, 192]

  padim_gram_kernel<<<dim3(PP), dim3(256), 0, stream>>>(
      emb, means_in, cov_in, out_means, out_cov);
}